// Loss_func_48387101557173
// MI455X (gfx1250) — compile-verified
//
#include <hip/hip_runtime.h>

// ---------------- problem constants (match reference) ----------------
#define BATCH   32
#define NA      3
#define NCLS    80
#define GRID    52
#define HW      (GRID*GRID)          // 2704
#define NC      (BATCH*NA*HW)        // 259584 dense cells
#define NT      50
#define ATTRS   85                   // 5 + 80
#define NBLK    (NC/256)             // 1014 dense-pass blocks (exact)
#define NBLK_PAD 1016                // padded to multiple of 4 for WMMA K-loop

typedef __attribute__((ext_vector_type(2))) float v2f;
typedef __attribute__((ext_vector_type(8))) float v8f;

__device__ __forceinline__ float sigm(float z) { return 1.f / (1.f + expf(-z)); }
// torch BCELoss clamp: clip(log(p), -100).  log(0) = -inf -> fmaxf gives -100.
__device__ __forceinline__ float clog(float p) { return fmaxf(logf(p), -100.f); }

// ---------------- kernel 1: init workspace ----------------
__global__ void init_ws(float* __restrict__ mask, float* __restrict__ noobj,
                        float* __restrict__ tx, float* __restrict__ ty,
                        float* __restrict__ tw, float* __restrict__ th,
                        unsigned* __restrict__ bits, float* __restrict__ partials) {
  int i = blockIdx.x * blockDim.x + threadIdx.x;
  if (i < NC) {
    mask[i] = 0.f; noobj[i] = 1.f;
    tx[i] = 0.f; ty[i] = 0.f; tw[i] = 0.f; th[i] = 0.f;
    bits[3*i] = 0u; bits[3*i+1] = 0u; bits[3*i+2] = 0u;
  }
  // zero the 2 pad columns of each of the 8 component rows (16 entries)
  if (i < 16)
    partials[(size_t)(i >> 1) * NBLK_PAD + NBLK + (i & 1)] = 0.f;
}

// ---------------- kernel 2: sparse target scatter ----------------
// One wave32; lane b handles batch image b, looping targets serially so the
// last valid target wins on duplicate cells (deterministic, matches scatter-set).
__global__ __launch_bounds__(32) void build_targets(
    const float* __restrict__ tgt, float* __restrict__ mask,
    float* __restrict__ noobj, float* __restrict__ tx, float* __restrict__ ty,
    float* __restrict__ tw, float* __restrict__ th, unsigned* __restrict__ bits) {
  int b = threadIdx.x;
  if (b >= BATCH) return;
  // anchors / stride (416/52 = 8)
  const float aw[3] = {10.f/8.f, 16.f/8.f, 33.f/8.f};
  const float ah[3] = {13.f/8.f, 30.f/8.f, 23.f/8.f};
  for (int t = 0; t < NT; ++t) {
    const float* p = tgt + ((size_t)b * NT + t) * 5;
    float c0 = p[0], x = p[1], y = p[2], w = p[3], h = p[4];
    if (c0 + x + y + w + h == 0.f) continue;            // skip empty target
    float gx = x * (float)GRID, gy = y * (float)GRID;
    float gw = w * (float)GRID, gh = h * (float)GRID;
    int gi = min(max((int)floorf(gx), 0), GRID - 1);
    int gj = min(max((int)floorf(gy), 0), GRID - 1);
    float iou[3]; float best = -1.f; int bn = 0;
    for (int a = 0; a < NA; ++a) {
      float inter = fminf(gw, aw[a]) * fminf(gh, ah[a]);
      float uni   = gw * gh + aw[a] * ah[a] - inter;
      iou[a] = inter / (uni + 1e-16f);
      if (iou[a] > best) { best = iou[a]; bn = a; }     // argmax = first max
    }
    int cell = ((b * NA + bn) * GRID + gj) * GRID + gi;
    mask[cell] = 1.f;
    tx[cell] = gx - (float)gi;
    ty[cell] = gy - (float)gj;
    tw[cell] = logf(gw / aw[bn] + 1e-16f);
    th[cell] = logf(gh / ah[bn] + 1e-16f);
    int ci = min(max((int)c0, 0), NCLS - 1);
    atomicOr(&bits[3*cell + (ci >> 5)], 1u << (ci & 31)); // scatter-max of one-hot
    // reference quirk: noobj[b, (iou>thr).long(), gj, gi] = 0  (anchor idx 0/1)
    for (int a = 0; a < NA; ++a) {
      int i01 = (iou[a] > 0.5f) ? 1 : 0;
      noobj[((b * NA + i01) * GRID + gj) * GRID + gi] = 0.f;
    }
  }
}

// ---------------- kernel 3: dense pass (only ~3.5 MB read, not 88 MB) ----------------
// acc: 0=sx 1=sy 2=sw 3=sh 4=sconf 5=snconf 6=scls 7=nsel
// partials stored COMPONENT-MAJOR: partials[c * NBLK_PAD + block]
__global__ __launch_bounds__(256) void dense_loss(
    const float* __restrict__ in, const float* __restrict__ mask,
    const float* __restrict__ noobj, const float* __restrict__ tx,
    const float* __restrict__ ty, const float* __restrict__ tw,
    const float* __restrict__ th, const unsigned* __restrict__ bits,
    float* __restrict__ partials) {
  int idx = blockIdx.x * blockDim.x + threadIdx.x;
  float acc[8] = {0.f,0.f,0.f,0.f,0.f,0.f,0.f,0.f};
  if (idx < NC) {
    int i = idx % GRID;
    int j = (idx / GRID) % GRID;
    int a = (idx / HW) % NA;
    int b = idx / (NA * HW);
    const float* base = in + ((size_t)b * (NA*ATTRS) + a * ATTRS) * HW + j * GRID + i;
    float conf = sigm(base[4 * HW]);
    if (noobj[idx] != 0.f)
      acc[5] = -clog(1.f - conf);                       // noobj conf BCE vs 0
    float m = mask[idx];
    if (m != 0.f) {                                     // ~0.5% of cells
      float px = sigm(base[0]);        float t = tx[idx];
      acc[0] = -(t * clog(px) + (1.f - t) * clog(1.f - px));
      float py = sigm(base[HW]);       t = ty[idx];
      acc[1] = -(t * clog(py) + (1.f - t) * clog(1.f - py));
      float dw = base[2 * HW] - tw[idx]; acc[2] = dw * dw;
      float dh = base[3 * HW] - th[idx]; acc[3] = dh * dh;
      acc[4] = -clog(conf);                             // conf BCE vs 1
      unsigned b0 = bits[3*idx], b1 = bits[3*idx+1], b2 = bits[3*idx+2];
      float s = 0.f;
      for (int c = 0; c < NCLS; ++c) {
        float pc = sigm(base[(5 + c) * HW]);
        unsigned wv = (c < 32) ? b0 : ((c < 64) ? b1 : b2);
        float tc = (float)((wv >> (c & 31)) & 1u);
        s += -(tc * clog(pc) + (1.f - tc) * clog(1.f - pc));
      }
      acc[6] = s;
      acc[7] = m;                                       // nsel
    }
  }
  __shared__ float red[256];
  for (int c = 0; c < 8; ++c) {
    red[threadIdx.x] = acc[c];
    __syncthreads();
    for (int s = 128; s > 0; s >>= 1) {
      if (threadIdx.x < s) red[threadIdx.x] += red[threadIdx.x + s];
      __syncthreads();
    }
    if (threadIdx.x == 0) partials[(size_t)c * NBLK_PAD + blockIdx.x] = red[0];
    __syncthreads();
  }
}

// ---------------- kernel 4: WMMA f32 reduction + final combine ----------------
// D = A(16x4) x ones(4x16) + C  =>  D[m][n] = sum_k A[m][k] + C[m][n].
// A f32 16x4 layout (ISA 7.12.2): lanes 0-15 hold {K0,K1}, lanes 16-31 hold {K2,K3};
// row m = loss component (0..7).  Component-major partials make each lane's
// (K, K+1) pair contiguous -> one unconditional global_load_b64 per iteration;
// the m<8 row mask is a branchless v_cndmask.  f32 WMMA => exact fp32 sums.
__global__ __launch_bounds__(32) void final_wmma(
    const float* __restrict__ partials, float* __restrict__ out) {
  int lane = threadIdx.x;
  int m = lane & 15;
  bool rowv = (m < 8);
  int kbase = (lane < 16) ? 0 : 2;
  const float* pbase = partials + (size_t)(m & 7) * NBLK_PAD + kbase;
  v8f c = {};
  v2f ones; ones.x = 1.f; ones.y = 1.f;
  for (int j = 0; j < NBLK_PAD; j += 4) {
    v2f av = *(const v2f*)(pbase + j);   // 8B-aligned pair (K, K+1)
    av.x = rowv ? av.x : 0.f;
    av.y = rowv ? av.y : 0.f;
    // 8 args: (neg_a, A, neg_b, B, c_mod, C, reuse_a, reuse_b)
    c = __builtin_amdgcn_wmma_f32_16x16x4_f32(false, av, false, ones,
                                              (short)0, c, false, false);
  }
  if (lane == 0) {   // C/D layout: VGPR r holds M=r, lane n holds N=n -> c[r] = D[r][0]
    const float invN = 1.f / (float)NC;
    float lx    = c[0] * invN;
    float ly    = c[1] * invN;
    float lw    = c[2] * invN;
    float lh    = c[3] * invN;
    float lconf = c[4] * invN;
    float lnconf = 0.5f * c[5] * invN;
    float lcls  = c[6] / fmaxf(c[7] * (float)NCLS, 1.f);
    float loss = 2.5f * (lx + ly) + 2.5f * (lw + lh) +
                 10.f * lconf + 3.f * lnconf + 20.f * lcls;
    out[0] = loss; out[1] = lx; out[2] = ly; out[3] = lw;
    out[4] = lh;   out[5] = lconf; out[6] = lcls;
  }
}

// ---------------- launcher ----------------
extern "C" void kernel_launch(void* const* d_in, const int* in_sizes, int n_in,
                              void* d_out, int out_size, void* d_ws, size_t ws_size,
                              hipStream_t stream) {
  (void)in_sizes; (void)n_in; (void)out_size; (void)ws_size;
  const float* inp = (const float*)d_in[0];   // [32,255,52,52] f32
  const float* tgt = (const float*)d_in[1];   // [32,50,5] f32
  float* out = (float*)d_out;                 // 7 f32 scalars

  float* f      = (float*)d_ws;
  float* mask   = f + 0 * (size_t)NC;
  float* noobj  = f + 1 * (size_t)NC;
  float* tx     = f + 2 * (size_t)NC;
  float* ty     = f + 3 * (size_t)NC;
  float* tw     = f + 4 * (size_t)NC;
  float* th     = f + 5 * (size_t)NC;
  unsigned* bits = (unsigned*)(f + 6 * (size_t)NC);    // 3*NC u32
  float* partials = f + 9 * (size_t)NC;                // 8 * NBLK_PAD f32

  init_ws<<<(NC + 255) / 256, 256, 0, stream>>>(mask, noobj, tx, ty, tw, th, bits, partials);
  build_targets<<<1, 32, 0, stream>>>(tgt, mask, noobj, tx, ty, tw, th, bits);
  dense_loss<<<NBLK, 256, 0, stream>>>(inp, mask, noobj, tx, ty, tw, th, bits, partials);
  final_wmma<<<1, 32, 0, stream>>>(partials, out);
}